// Corr3D_26396869001363
// MI455X (gfx1250) — compile-verified
//
#include <hip/hip_runtime.h>

// Corr3D (3x3x3 local correlation) for MI455X / gfx1250.
// Gram-band WMMA formulation: S[t,t'] = sum_c q[c,t] * k[c,t'] via chained
// V_WMMA_F32_16X16X4_F32 (K=4 x 8 steps = C=32); the 3 diagonals of S are the
// dt taps. 14 useful rows per 16-row tile (T = 112 = 8*14).

typedef __attribute__((ext_vector_type(2))) float v2f;
typedef __attribute__((ext_vector_type(8))) float v8f;

namespace {
constexpr int B_ = 2;
constexpr int C_ = 32;
constexpr int H_ = 80;
constexpr int W_ = 96;
constexpr int T_ = 112;
constexpr int TT = 14;            // output t's per tile (rows 0..13 of S used)
constexpr int NTT = T_ / TT;      // 8 tiles, exact
constexpr int WAVES_PER_BLOCK = 4;

constexpr int  sW  = T_;                     // 112
constexpr int  sH  = W_ * T_;                // 10752
constexpr long sC  = (long)H_ * W_ * T_;     // 860160
constexpr long sB  = (long)C_ * sC;          // q/k batch stride
constexpr long sTap = sC;                    // out tap stride
constexpr long sBo  = 27 * sC;               // out batch stride
}

__global__ __launch_bounds__(32 * WAVES_PER_BLOCK)
void corr3d_wmma_kernel(const float* __restrict__ q,
                        const float* __restrict__ k,
                        float* __restrict__ out) {
  __shared__ float smem[WAVES_PER_BLOCK * 16 * 16];

  const int lane   = threadIdx.x & 31;
  const int wave   = threadIdx.x >> 5;
  const int lane16 = lane & 15;
  const int chalf  = (lane >> 4) << 1;   // 0 for lanes 0-15, 2 for lanes 16-31

  // One wave == one (b, h, w, t-tile).
  int id = blockIdx.x * WAVES_PER_BLOCK + wave;
  const int tt = id % NTT; id /= NTT;
  const int w  = id % W_;  id /= W_;
  const int h  = id % H_;  id /= H_;
  const int b  = id;       // < B_
  const int t0 = tt * TT;

  // ---- A matrix (16x4 per step): row m = lane16 -> t = t0 + lane16.
  // A VGPR layout: lanes 0-15 hold K = chalf..chalf+1 (c within 4-chunk).
  // Scale q by 1/C here; rows 14,15 may run past T on the last tile -> zero.
  const int  ta   = t0 + lane16;
  const bool taOK = (ta < T_);
  const long qbase = (long)b * sB + (long)h * sH + (long)w * sW + ta;
  const float scale = 1.0f / (float)C_;
  v2f a[8];
#pragma unroll
  for (int cc = 0; cc < 8; ++cc) {
    const long off = qbase + (long)(cc * 4 + chalf) * sC;
    const float x0 = q[taOK ? off      : 0];
    const float x1 = q[taOK ? off + sC : 0];
    a[cc].x = taOK ? x0 * scale : 0.0f;
    a[cc].y = taOK ? x1 * scale : 0.0f;
  }

  // ---- B matrix column: t' = t0 - 1 + lane16 (covers t-1..t+1 for rows 0..13)
  const int  tp   = t0 - 1 + lane16;
  const bool tpOK = (tp >= 0) && (tp < T_);
  float* S = &smem[wave * 256];

  for (int dh = 0; dh < 3; ++dh) {
    const int hh = h + dh - 1;
    for (int dw = 0; dw < 3; ++dw) {
      const int  ww   = w + dw - 1;
      const bool bOK  = tpOK && (hh >= 0) && (hh < H_) && (ww >= 0) && (ww < W_);
      const long kbase = (long)b * sB + (long)hh * sH + (long)ww * sW + tp;

      v8f acc = {0.f, 0.f, 0.f, 0.f, 0.f, 0.f, 0.f, 0.f};
#pragma unroll
      for (int cc = 0; cc < 8; ++cc) {
        const long off = kbase + (long)(cc * 4 + chalf) * sC;
        const float y0 = k[bOK ? off      : 0];
        const float y1 = k[bOK ? off + sC : 0];
        v2f bb;
        bb.x = bOK ? y0 : 0.0f;
        bb.y = bOK ? y1 : 0.0f;
        // acc(16x16,f32) += A(16x4,f32) x B(4x16,f32)
        acc = __builtin_amdgcn_wmma_f32_16x16x4_f32(
            false, a[cc], false, bb, (short)0, acc, false, false);
      }

      // Spill S tile: VGPR r holds M = r + 8*(lane>=16), N = lane16.
#pragma unroll
      for (int r = 0; r < 8; ++r) {
        const int m = r + ((lane >> 4) << 3);
        S[m * 16 + lane16] = acc[r];
      }
      __syncthreads();

      // Diagonals: out[tap=(dh,dw,dt)][t0+r] = S[r][r+dt], r = 0..13.
      if (lane < TT) {
        const int  r = lane;
        const long obase = (long)b * sBo + (long)h * sH + (long)w * sW + (t0 + r);
#pragma unroll
        for (int dt = 0; dt < 3; ++dt) {
          const int tap = dh * 9 + dw * 3 + dt;
          out[obase + (long)tap * sTap] = S[r * 16 + r + dt];
        }
      }
      __syncthreads();  // protect S against next iteration's writes
    }
  }
}

extern "C" void kernel_launch(void* const* d_in, const int* in_sizes, int n_in,
                              void* d_out, int out_size, void* d_ws, size_t ws_size,
                              hipStream_t stream) {
  (void)in_sizes; (void)n_in; (void)out_size; (void)d_ws; (void)ws_size;
  const float* q = (const float*)d_in[0];
  const float* k = (const float*)d_in[1];
  float* out = (float*)d_out;

  constexpr int total_waves = B_ * H_ * W_ * NTT;           // 122880
  constexpr int blocks = total_waves / WAVES_PER_BLOCK;     // 30720
  corr3d_wmma_kernel<<<blocks, 32 * WAVES_PER_BLOCK, 0, stream>>>(q, k, out);
}